// STraTS_text_10565619548197
// MI455X (gfx1250) — compile-verified
//
#include <hip/hip_runtime.h>

// ---------------------------------------------------------------------------
// STraTS-style masked multi-head attention, fully WMMA-based (fp32 WMMA).
// B=16, LQ=64, LK=256, EMBED=128, H=8, DK=16, DIM=48, NHIDDEN=128.
//
// Softmax simplification (exact in fp32 arithmetic): with mask value -10000,
// exp(-10000 - c) == 0.0f, so per (b,h,q):
//   e_k   = exp(s_k - max_k s_k)
//   num_d = sum_k e_k * mask[k,d] * value[k,d]
//   den_d = sum_k e_k * mask[k,d]
//   x_d   = num_d / den_d
// i.e. the masked-softmax contraction is a GEMM E(64x256) @ CM(256x96).
//
// The 1/sqrt(DK)=0.25 score scale is folded into the Q projection.
// ---------------------------------------------------------------------------

typedef float v2f __attribute__((ext_vector_type(2)));
typedef float v8f __attribute__((ext_vector_type(8)));

#define B_    16
#define LQ_   64
#define LK_   256
#define ET_   128
#define H_    8
#define DK_   16
#define DIM_  48
#define NH_   128

// D(16x16,f32) = A(16x4,f32) x B(4x16,f32) + C
// 8-arg form: (neg_a, A, neg_b, B, c_mod, C, reuse_a, reuse_b)
static __device__ __forceinline__ v8f wmma_f32(v2f a, v2f b, v8f c) {
  return __builtin_amdgcn_wmma_f32_16x16x4_f32(false, a, false, b, (short)0, c,
                                               false, false);
}

// ---------------------------------------------------------------------------
// Kernel 1/2: Y[m][n] = (sum_k X[m][k]*W[n][k] + bias[n]) * SCALEQ, head-split:
//   Y[((batch*H + head)*L + row_in_batch)*DK + dk],  head = n>>4, dk = n&15.
// One wave computes one 16x16 output tile; K=128 -> 32 WMMA per tile.
// L is a compile-time power of two so the batch split is shift/and.
// ---------------------------------------------------------------------------
template <int L, bool SCALEQ>
__global__ __launch_bounds__(128) void proj_kernel(
    const float* __restrict__ X, const float* __restrict__ W,
    const float* __restrict__ bias, float* __restrict__ Y) {
  const int lane = threadIdx.x & 31;
  const int wave = threadIdx.x >> 5;
  const int tile = blockIdx.x * 4 + wave;
  const int tm = tile >> 3;  // 16-row M tile
  const int tn = tile & 7;   // 16-col N tile == head index
  const int mn = lane & 15;
  const int kh = lane >> 4;

  const float* arow = X + (tm * 16 + mn) * ET_;
  const float* brow = W + (tn * 16 + mn) * ET_;
  v8f c = {};
#pragma unroll 4
  for (int k0 = 0; k0 < ET_; k0 += 4) {
    v2f a = *(const v2f*)(arow + k0 + 2 * kh);
    v2f b = *(const v2f*)(brow + k0 + 2 * kh);
    c = wmma_f32(a, b, c);
  }
  const float bn = bias[tn * 16 + mn];
#pragma unroll
  for (int r = 0; r < 8; ++r) {
    unsigned mrow = tm * 16 + r + 8 * kh;  // C/D row = r + 8*(lane>>4)
    unsigned bidx = mrow / L;              // L is constexpr pow2 -> shift
    unsigned lrow = mrow & (L - 1);
    float y = c[r] + bn;
    if (SCALEQ) y *= 0.25f;  // fold 1/sqrt(DK) into q
    Y[((bidx * H_ + tn) * L + lrow) * DK_ + mn] = y;
  }
}

// ---------------------------------------------------------------------------
// Kernel 3: build CM[b][k][0..47] = mask*value, CM[b][k][48..95] = mask
// ---------------------------------------------------------------------------
__global__ __launch_bounds__(256) void cm_kernel(const float* __restrict__ value,
                                                 const int* __restrict__ mask,
                                                 float* __restrict__ CM) {
  int i = blockIdx.x * 256 + threadIdx.x;  // over B*LK*DIM = 196608
  if (i < B_ * LK_ * DIM_) {
    int d = i % DIM_;
    int bk = i / DIM_;  // b*LK + k
    float m = (mask[i] != 0) ? 1.0f : 0.0f;
    float* row = CM + bk * 96;
    row[d] = m * value[i];
    row[48 + d] = m;
  }
}

// ---------------------------------------------------------------------------
// Kernel 4: fused attention per (b,h) block.  256 threads = 8 waves.
// LDS: E[64][256] (exp'd scores) + ND[64][96] (num|den, reused as scratch).
// ---------------------------------------------------------------------------
__global__ __launch_bounds__(256) void attn_kernel(
    const float* __restrict__ Qp, const float* __restrict__ Kp,
    const float* __restrict__ CM, float* __restrict__ xbuf) {
  extern __shared__ float lds[];
  float* E = lds;               // 64*256 floats
  float* ND = lds + LQ_ * LK_;  // 64*96 floats (scratch during phase 2)

  const int bh = blockIdx.x;
  const int b = bh >> 3;
  const int h = bh & 7;
  const int t = threadIdx.x;
  const int lane = t & 31;
  const int wave = t >> 5;
  const int mn = lane & 15;
  const int kh = lane >> 4;

  const float* Qb = Qp + bh * LQ_ * DK_;
  const float* Kb = Kp + bh * LK_ * DK_;
  const float* CMb = CM + b * LK_ * 96;

  // ---- Phase 1: S = (q/4) Kt into E.  64 tiles, 8 per wave, K=16.
  for (int i = 0; i < 8; ++i) {
    int tile = wave * 8 + i;
    int tm = tile >> 4;  // 0..3
    int tn = tile & 15;  // 0..15
    v8f c = {};
#pragma unroll
    for (int k0 = 0; k0 < DK_; k0 += 4) {
      v2f a = *(const v2f*)(Qb + (tm * 16 + mn) * DK_ + k0 + 2 * kh);
      v2f bb = *(const v2f*)(Kb + (tn * 16 + mn) * DK_ + k0 + 2 * kh);
      c = wmma_f32(a, bb, c);
    }
#pragma unroll
    for (int r = 0; r < 8; ++r)
      E[(tm * 16 + r + 8 * kh) * LK_ + tn * 16 + mn] = c[r];
  }
  __syncthreads();

  // ---- Phase 2: row max over LK, then E = exp(E - rowmax).
  {
    int row = t >> 2, seg = t & 3;
    float mx = -3.4e38f;
    const float* er = E + row * LK_ + seg * 64;
    for (int j = 0; j < 64; ++j) mx = fmaxf(mx, er[j]);
    ND[row * 4 + seg] = mx;  // partials in scratch
  }
  __syncthreads();
  float* rowmax = ND + 256;
  if (t < 64) {
    rowmax[t] = fmaxf(fmaxf(ND[t * 4], ND[t * 4 + 1]),
                      fmaxf(ND[t * 4 + 2], ND[t * 4 + 3]));
  }
  __syncthreads();
  for (int idx = t; idx < LQ_ * LK_; idx += 256) {
    int r = idx >> 8;
    E[idx] = __expf(E[idx] - rowmax[r]);
  }
  __syncthreads();

  // ---- Phase 3: [num|den](64x96) = E(64x256) @ CMb(256x96).  24 tiles, K=256.
  for (int i = 0; i < 3; ++i) {
    int tile = wave * 3 + i;
    int tm = tile / 6;  // 0..3
    int tn = tile % 6;  // 0..5
    v8f c = {};
#pragma unroll 4
    for (int k0 = 0; k0 < LK_; k0 += 4) {
      v2f a = *(const v2f*)(E + (tm * 16 + mn) * LK_ + k0 + 2 * kh);
      v2f bb;
      bb.x = CMb[(k0 + 2 * kh) * 96 + tn * 16 + mn];
      bb.y = CMb[(k0 + 2 * kh + 1) * 96 + tn * 16 + mn];
      c = wmma_f32(a, bb, c);
    }
#pragma unroll
    for (int r = 0; r < 8; ++r)
      ND[(tm * 16 + r + 8 * kh) * 96 + tn * 16 + mn] = c[r];
  }
  __syncthreads();

  // ---- Phase 4: x = num/den, scatter to (B, LQ, H*DIM) layout.
  for (int idx = t; idx < LQ_ * DIM_; idx += 256) {
    int m = idx / DIM_;
    int d = idx - m * DIM_;
    float num = ND[m * 96 + d];
    float den = ND[m * 96 + 48 + d];
    xbuf[(b * LQ_ + m) * (H_ * DIM_) + h * DIM_ + d] = num / den;
  }
}

// ---------------------------------------------------------------------------
// Kernel 5: out = Xb(1024x384) @ Wo^T(384x128) + bo.  K=384 -> 96 WMMA/tile.
// ---------------------------------------------------------------------------
__global__ __launch_bounds__(128) void outproj_kernel(
    const float* __restrict__ Xb, const float* __restrict__ Wo,
    const float* __restrict__ bo, float* __restrict__ out) {
  const int lane = threadIdx.x & 31;
  const int wave = threadIdx.x >> 5;
  const int tile = blockIdx.x * 4 + wave;
  const int tm = tile >> 3;  // 0..63
  const int tn = tile & 7;   // 0..7
  const int mn = lane & 15;
  const int kh = lane >> 4;

  const float* arow = Xb + (tm * 16 + mn) * (H_ * DIM_);
  const float* brow = Wo + (tn * 16 + mn) * (H_ * DIM_);
  v8f c = {};
#pragma unroll 4
  for (int k0 = 0; k0 < H_ * DIM_; k0 += 4) {
    v2f a = *(const v2f*)(arow + k0 + 2 * kh);
    v2f b = *(const v2f*)(brow + k0 + 2 * kh);
    c = wmma_f32(a, b, c);
  }
  const float bn = bo[tn * 16 + mn];
#pragma unroll
  for (int r = 0; r < 8; ++r)
    out[(tm * 16 + r + 8 * kh) * NH_ + tn * 16 + mn] = c[r] + bn;
}

// ---------------------------------------------------------------------------
extern "C" void kernel_launch(void* const* d_in, const int* in_sizes, int n_in,
                              void* d_out, int out_size, void* d_ws,
                              size_t ws_size, hipStream_t stream) {
  const float* query = (const float*)d_in[0];
  const float* key   = (const float*)d_in[1];
  const float* value = (const float*)d_in[2];
  const int*   mask  = (const int*)d_in[3];
  const float* Wq = (const float*)d_in[4];
  const float* bq = (const float*)d_in[5];
  const float* Wk = (const float*)d_in[6];
  const float* bk = (const float*)d_in[7];
  const float* Wo = (const float*)d_in[8];
  const float* bo = (const float*)d_in[9];
  float* out = (float*)d_out;

  float* ws = (float*)d_ws;
  float* Qp = ws;                         // B*H*LQ*DK = 131072 floats
  float* Kp = Qp + B_ * H_ * LQ_ * DK_;   // B*H*LK*DK = 524288 floats
  float* CM = Kp + B_ * H_ * LK_ * DK_;   // B*LK*96   = 393216 floats
  float* Xb = CM + B_ * LK_ * 96;         // B*LQ*384  = 393216 floats

  // Q projection (scores scale folded in): 512 tiles -> 128 blocks of 4 waves.
  proj_kernel<LQ_, true><<<128, 128, 0, stream>>>(query, Wq, bq, Qp);
  // K projection: 2048 tiles -> 512 blocks.
  proj_kernel<LK_, false><<<512, 128, 0, stream>>>(key, Wk, bk, Kp);
  // mask*value | mask prep.
  cm_kernel<<<768, 256, 0, stream>>>(value, mask, CM);
  // fused attention, one block per (b,h); 88KB dynamic LDS.
  attn_kernel<<<B_ * H_, 256, (LQ_ * LK_ + LQ_ * 96) * sizeof(float), stream>>>(
      Qp, Kp, CM, Xb);
  // output projection: 1024x384 @ 384x128.
  outproj_kernel<<<128, 128, 0, stream>>>(Xb, Wo, bo, out);
}